// Temporal_Attention_14920716386674
// MI455X (gfx1250) — compile-verified
//
#include <hip/hip_runtime.h>
#include <hip/hip_bf16.h>

// ---------------------------------------------------------------------------
// Problem constants (fixed by the reference setup_inputs)
// ---------------------------------------------------------------------------
#define BATCH   2
#define HH      64
#define WW      64
#define INC     128
#define OUTC    128
#define GROUPS  8
#define CG      16          // channels per group
#define KWIN    7           // 2D window
#define PADW    3
#define RWIN    15          // refine window 2*K+1
#define NPIX    (BATCH*HH*WW)     // 8192

typedef float v2f __attribute__((ext_vector_type(2)));
typedef float v8f __attribute__((ext_vector_type(8)));

// ---------------------------------------------------------------------------
// Phase 1: fused q/k/v 1x1-conv GEMMs via V_WMMA_F32_16X16X4_F32.
//   out[p][o] = sum_c W[o][c] * X[c][p]   (pixel-major outputs for phase 2)
// One block = 16-pixel tile (contiguous in w), 8 waves = 8 out-channel stripes.
// ---------------------------------------------------------------------------
__global__ __launch_bounds__(256)
void qkv_gemm_kernel(const float* __restrict__ fm,
                     const float* __restrict__ wq,
                     const float* __restrict__ wk,
                     const float* __restrict__ wv,
                     float* __restrict__ q_ws,
                     float* __restrict__ k_ws,
                     float* __restrict__ v_ws)
{
    __shared__ float xt[2][INC][16];            // [t0/t1][in-ch][pixel]   16 KB

    const int tid = threadIdx.x;
    const int p0  = blockIdx.x * 16;            // flat pixel base (b*4096+h*64+w)
    const int b   = p0 >> 12;
    const int hw  = p0 & 4095;

    // ---- stage activation tiles into LDS (coalesced 64B rows) ----
    {
        const int px = tid & 15;
        const int c0 = tid >> 4;                // 0..15
        const float* b0 = fm + ((size_t)b * (2*INC)      ) * (HH*WW) + hw + px; // t0
        const float* b1 = fm + ((size_t)b * (2*INC) + INC) * (HH*WW) + hw + px; // t1
        #pragma unroll
        for (int r = 0; r < 8; ++r) {
            const int c = c0 + 16 * r;
            xt[0][c][px] = b0[(size_t)c * (HH*WW)];
            xt[1][c][px] = b1[(size_t)c * (HH*WW)];
        }
    }
    __syncthreads();

    const int lane = tid & 31;
    const int wave = tid >> 5;                  // 0..7 -> out-channel stripe
    const int n    = lane & 15;                 // A row (pixel) and B col (out ch)
    const int hi   = lane >> 4;                 // 0/1
    const int koff = hi * 2;                    // K sub-offset per ISA A/B layout
    const int o0   = wave * 16;

    const float* Wmat[3] = { wq, wk, wv };
    float*       Dmat[3] = { q_ws, k_ws, v_ws };

    #pragma unroll
    for (int m = 0; m < 3; ++m) {
        const int src     = (m == 0) ? 1 : 0;   // q reads fm_t1; k,v read fm_t0
        const float* W    = Wmat[m];
        float*       D    = Dmat[m];

        v8f acc0 = {0.f,0.f,0.f,0.f,0.f,0.f,0.f,0.f};
        v8f acc1 = {0.f,0.f,0.f,0.f,0.f,0.f,0.f,0.f};

        #pragma unroll
        for (int kb = 0; kb < INC; kb += 4) {
            v2f a;                               // A: 16x4 fp32 tile
            a.x = xt[src][kb + koff + 0][n];
            a.y = xt[src][kb + koff + 1][n];
            v2f bv = *(const v2f*)(W + (size_t)(o0 + n) * INC + kb + koff);
            if (kb < INC/2)
                acc0 = __builtin_amdgcn_wmma_f32_16x16x4_f32(
                        false, a, false, bv, (short)0, acc0, false, false);
            else
                acc1 = __builtin_amdgcn_wmma_f32_16x16x4_f32(
                        false, a, false, bv, (short)0, acc1, false, false);
        }
        const v8f acc = acc0 + acc1;

        // D tile: VGPR r -> pixel (p0 + r + 8*hi), out-channel (o0 + n)
        #pragma unroll
        for (int r = 0; r < 8; ++r)
            D[(size_t)(p0 + r + 8*hi) * OUTC + o0 + n] = acc[r];
    }
}

// ---------------------------------------------------------------------------
// Phase 2: windowed attention + 1D refinements.
// One wave = (group, 2 pixels); lanes = (pixel-half, channel 0..15).
// 16-wide dot products via shfl_xor butterfly inside each 16-lane half.
// ---------------------------------------------------------------------------
__device__ __forceinline__ float half_reduce(float v) {
    v += __shfl_xor(v, 8, 32);
    v += __shfl_xor(v, 4, 32);
    v += __shfl_xor(v, 2, 32);
    v += __shfl_xor(v, 1, 32);
    return v;                                   // broadcast within 16-lane half
}

__global__ __launch_bounds__(256)
void attn_kernel(const float* __restrict__ q_ws,
                 const float* __restrict__ k_ws,
                 const float* __restrict__ v_ws,
                 const float* __restrict__ rel_h,
                 const float* __restrict__ rel_w,
                 float* __restrict__ out)
{
    const int tid  = threadIdx.x;
    const int wave = tid >> 5;
    const int lane = tid & 31;
    const int c    = lane & 15;
    const int pp   = lane >> 4;

    const int task = blockIdx.x * 8 + wave;     // 32768 tasks = 8 groups * 4096 pairs
    const int g    = task >> 12;
    const int pair = task & 4095;
    const int p    = pair * 2 + pp;             // flat pixel
    const int b    = p >> 12;
    const int hw   = p & 4095;
    const int h    = hw >> 6;
    const int w    = hw & 63;
    const int gc   = g * CG + c;

    const size_t bbase = (size_t)b << 12;       // b*4096
    const float  q     = q_ws[(size_t)p * OUTC + gc];

    // rel bias: g<4 -> bias over i (rel_h), g>=4 -> bias over j (rel_w)
    const float* rel = (g < 4) ? (rel_h + (size_t)gc * KWIN)
                               : (rel_w + (size_t)(gc - 64) * KWIN);
    float bias[KWIN];
    #pragma unroll
    for (int t = 0; t < KWIN; ++t)
        bias[t] = half_reduce(q * rel[t]);

    // -------------------- 2D 7x7 attention --------------------
    float lg[KWIN * KWIN];
    float mx = -3.0e38f;
    #pragma unroll
    for (int i = 0; i < KWIN; ++i) {
        const int  hh = h + i - PADW;
        const bool vi = (unsigned)hh < (unsigned)HH;
        #pragma unroll
        for (int j = 0; j < KWIN; ++j) {
            const int ww = w + j - PADW;
            float d = 0.f;
            if (vi && (unsigned)ww < (unsigned)WW)
                d = q * k_ws[(bbase + (hh << 6) + ww) * OUTC + gc];
            d = half_reduce(d);
            d += (g < 4) ? bias[i] : bias[j];   // rel applies even at OOB
            lg[i * KWIN + j] = d;
            mx = fmaxf(mx, d);
        }
    }
    float s = 0.f, acc = 0.f;
    #pragma unroll
    for (int i = 0; i < KWIN; ++i) {
        const int hh = h + i - PADW;
        #pragma unroll
        for (int j = 0; j < KWIN; ++j) {
            const int  ww = w + j - PADW;
            const float e = __expf(lg[i * KWIN + j] - mx);
            s += e;
            if ((unsigned)hh < (unsigned)HH && (unsigned)ww < (unsigned)WW)
                acc += e * v_ws[(bbase + (hh << 6) + ww) * OUTC + gc];
        }
    }
    float result = acc / s;

    // -------------------- row refinement (along w, win=15, no bias) ------
    {
        float lr[RWIN];
        float m2 = -3.0e38f;
        #pragma unroll
        for (int j = 0; j < RWIN; ++j) {
            const int ww = w + j - KWIN;
            float d = 0.f;
            if ((unsigned)ww < (unsigned)WW)
                d = q * k_ws[(bbase + (h << 6) + ww) * OUTC + gc];
            d = half_reduce(d);                 // OOB logit is exactly 0
            lr[j] = d;
            m2 = fmaxf(m2, d);
        }
        float ss = 0.f, aa = 0.f;
        #pragma unroll
        for (int j = 0; j < RWIN; ++j) {
            const int  ww = w + j - KWIN;
            const float e = __expf(lr[j] - m2);
            ss += e;
            if ((unsigned)ww < (unsigned)WW)
                aa += e * v_ws[(bbase + (h << 6) + ww) * OUTC + gc];
        }
        result += aa / ss;
    }

    // -------------------- col refinement (along h) -----------------------
    {
        float lc[RWIN];
        float m2 = -3.0e38f;
        #pragma unroll
        for (int j = 0; j < RWIN; ++j) {
            const int hh = h + j - KWIN;
            float d = 0.f;
            if ((unsigned)hh < (unsigned)HH)
                d = q * k_ws[(bbase + (hh << 6) + w) * OUTC + gc];
            d = half_reduce(d);
            lc[j] = d;
            m2 = fmaxf(m2, d);
        }
        float ss = 0.f, aa = 0.f;
        #pragma unroll
        for (int j = 0; j < RWIN; ++j) {
            const int  hh = h + j - KWIN;
            const float e = __expf(lc[j] - m2);
            ss += e;
            if ((unsigned)hh < (unsigned)HH)
                aa += e * v_ws[(bbase + (hh << 6) + w) * OUTC + gc];
        }
        result += aa / ss;
    }

    // out layout: (B, OUTC, H, W)
    out[((size_t)(b * OUTC + gc) << 12) + (h << 6) + w] = result;
}

// ---------------------------------------------------------------------------
extern "C" void kernel_launch(void* const* d_in, const int* in_sizes, int n_in,
                              void* d_out, int out_size, void* d_ws, size_t ws_size,
                              hipStream_t stream)
{
    const float* fm    = (const float*)d_in[0];
    const float* wq    = (const float*)d_in[1];
    const float* wk    = (const float*)d_in[2];
    const float* wv    = (const float*)d_in[3];
    const float* rel_h = (const float*)d_in[4];
    const float* rel_w = (const float*)d_in[5];
    float*       out   = (float*)d_out;

    float* ws   = (float*)d_ws;
    float* q_ws = ws;                           // [NPIX][OUTC] pixel-major
    float* k_ws = ws + (size_t)NPIX * OUTC;
    float* v_ws = ws + (size_t)2 * NPIX * OUTC;

    qkv_gemm_kernel<<<NPIX / 16, 256, 0, stream>>>(fm, wq, wk, wv, q_ws, k_ws, v_ws);
    attn_kernel<<<(GROUPS * NPIX / 2) / 8, 256, 0, stream>>>(q_ws, k_ws, v_ws,
                                                             rel_h, rel_w, out);
}